// MMDLoss_87806311399613
// MI455X (gfx1250) — compile-verified
//
#include <hip/hip_runtime.h>
#include <hip/hip_bf16.h>

typedef __attribute__((ext_vector_type(2))) float v2f;
typedef __attribute__((ext_vector_type(8))) float v8f;

constexpr int N  = 8192;
constexpr int D  = 256;
constexpr int BM = 64;    // block tile rows (i)
constexpr int BN = 32;    // block tile cols (j)
constexpr int KC = 64;    // K chunk staged in LDS
constexpr int LDA = KC + 4; // 68 dwords: keeps float4 stores 16B-aligned, stride%64==4 -> conflict-free frags

// ---------------------------------------------------------------- utilities
__global__ void initk(double* sumL2, double* S, int* n0) {
    *sumL2 = 0.0; *S = 0.0; *n0 = 0;
}

// one wave per row: sq[i] = sum_k X[i,k]^2 ; also count n0 = #(sub==0)
__global__ __launch_bounds__(256) void rownorm(const float* __restrict__ X,
                                               const int* __restrict__ sub,
                                               float* __restrict__ sq,
                                               int* __restrict__ n0) {
    const int lane = threadIdx.x & 31;
    const int wid  = threadIdx.x >> 5;
    const int row  = blockIdx.x * 8 + wid;
    const float* xr = X + (size_t)row * D;
    float4 v0 = *(const float4*)&xr[lane * 4];
    float4 v1 = *(const float4*)&xr[128 + lane * 4];
    float s = v0.x*v0.x + v0.y*v0.y + v0.z*v0.z + v0.w*v0.w
            + v1.x*v1.x + v1.y*v1.y + v1.z*v1.z + v1.w*v1.w;
#pragma unroll
    for (int off = 16; off > 0; off >>= 1) s += __shfl_down(s, off, 32);
    if (lane == 0) {
        sq[row] = s;
        if (sub[row] == 0) atomicAdd(n0, 1);
    }
}

__global__ void weightsk(const int* __restrict__ sub, const int* __restrict__ n0p,
                         float* __restrict__ w) {
    const int i = blockIdx.x * 256 + threadIdx.x;
    const int n0 = *n0p;
    const int n1 = N - n0;
    w[i] = (sub[i] == 0) ? (1.0f / (float)n0) : (-1.0f / (float)n1);
}

__global__ void bwk(const double* __restrict__ sumL2, float* __restrict__ ibs) {
    double bw = (*sumL2) / ((double)N * (double)N - (double)N);
    bw = bw / 4.0;   // KERNEL_MUL^(KERNEL_NUM/2) = 2^2
    double m = 1.0;
    for (int q = 0; q < 5; ++q) { ibs[q] = (float)(1.0 / (bw * m)); m *= 2.0; }
}

__global__ void fin(const double* __restrict__ S, float* __restrict__ out) {
    out[0] = (float)(*S);
}

// ---------------------------------------------------------------- fused GEMM passes
// PASS2=false : acc += sum over tile of max(sq_i + sq_j - 2*G_ij, 0)
// PASS2=true  : acc += sum over tile of w_i * w_j * sum_q exp(-L2 * ibs[q])
template <bool PASS2>
__global__ __launch_bounds__(256) void mmd_pass(const float* __restrict__ X,
                                                const float* __restrict__ sq,
                                                const float* __restrict__ wv,
                                                const float* __restrict__ ibs,
                                                double* __restrict__ acc) {
    __shared__ __align__(16) float As[BM][LDA];
    __shared__ __align__(16) float Bs[BN][LDA];
    __shared__ float sqA[BM], sqB[BN], wA[BM], wB[BN];
    __shared__ float red[8];

    const int tid  = threadIdx.x;
    const int lane = tid & 31;
    const int wid  = tid >> 5;        // 8 waves
    const int mi   = wid & 3;         // 4 wave-rows
    const int nj   = wid >> 2;        // 2 wave-cols
    const int half = lane >> 4;       // lanes 0-15 / 16-31
    const int lr   = lane & 15;
    const int i0   = blockIdx.y * BM;
    const int j0   = blockIdx.x * BN;

    // stage per-row norms / weights for the tile
    if (tid < BM) {
        sqA[tid] = sq[i0 + tid];
        if (PASS2) wA[tid] = wv[i0 + tid];
    } else if (tid < BM + BN) {
        const int r = tid - BM;
        sqB[r] = sq[j0 + r];
        if (PASS2) wB[r] = wv[j0 + r];
    }

    float ib[5];
    if (PASS2) {
#pragma unroll
        for (int q = 0; q < 5; ++q) ib[q] = ibs[q];
    }

    v8f c = {};  // 16x16 f32 accumulator, zero-init

    const int col4 = tid & 15;   // 16 float4 per KC-chunk row
    const int row  = tid >> 4;   // 0..15

    for (int kc = 0; kc < D; kc += KC) {
        __syncthreads();   // also covers sqA/sqB staging on first trip
        // cooperative load: A tile 64 x KC, B tile 32 x KC (float4, coalesced)
#pragma unroll
        for (int q = 0; q < 4; ++q) {
            const int r = row + q * 16;
            float4 v = *(const float4*)&X[(size_t)(i0 + r) * D + kc + col4 * 4];
            *(float4*)&As[r][col4 * 4] = v;
        }
#pragma unroll
        for (int q = 0; q < 2; ++q) {
            const int r = row + q * 16;
            float4 v = *(const float4*)&X[(size_t)(j0 + r) * D + kc + col4 * 4];
            *(float4*)&Bs[r][col4 * 4] = v;
        }
        __syncthreads();

        // V_WMMA_F32_16X16X4_F32 fragment layout (ISA 7.12.2, 32-bit A 16x4):
        //   lanes 0-15 : M=lane,    VGPR0=K0, VGPR1=K1
        //   lanes16-31 : M=lane-16, VGPR0=K2, VGPR1=K3
        // B (4x16) mirrors: lane holds N=lr, VGPR0/1 hold K = half*2 + {0,1}
#pragma unroll
        for (int kk = 0; kk < KC / 4; ++kk) {
            const int kb = kk * 4 + half * 2;
            v2f a, b;
            a.x = As[mi * 16 + lr][kb];
            a.y = As[mi * 16 + lr][kb + 1];
            b.x = Bs[nj * 16 + lr][kb];
            b.y = Bs[nj * 16 + lr][kb + 1];
            c = __builtin_amdgcn_wmma_f32_16x16x4_f32(
                    false, a, false, b, (short)0, c, false, false);
        }
    }

    // epilogue: C layout — VGPR r: lanes0-15 -> M=r, lanes16-31 -> M=r+8; N=lr
    const int ibase = mi * 16 + half * 8;
    const int jj    = nj * 16 + lr;
    const float sqj = sqB[jj];
    float lsum = 0.0f;
#pragma unroll
    for (int r = 0; r < 8; ++r) {
        const float g  = c[r];
        float l2 = sqA[ibase + r] + sqj - 2.0f * g;
        l2 = fmaxf(l2, 0.0f);
        if (PASS2) {
            float k = __expf(-l2 * ib[0]) + __expf(-l2 * ib[1]) + __expf(-l2 * ib[2])
                    + __expf(-l2 * ib[3]) + __expf(-l2 * ib[4]);
            lsum += wA[ibase + r] * wB[jj] * k;
        } else {
            lsum += l2;
        }
    }
#pragma unroll
    for (int off = 16; off > 0; off >>= 1) lsum += __shfl_down(lsum, off, 32);
    if (lane == 0) red[wid] = lsum;
    __syncthreads();
    if (tid == 0) {
        float bs = 0.0f;
#pragma unroll
        for (int wq = 0; wq < 8; ++wq) bs += red[wq];
        atomicAdd(acc, (double)bs);
    }
}

// ---------------------------------------------------------------- launch
extern "C" void kernel_launch(void* const* d_in, const int* in_sizes, int n_in,
                              void* d_out, int out_size, void* d_ws, size_t ws_size,
                              hipStream_t stream) {
    const int*   sub = (const int*)d_in[0];     // subggroup (N,1) int
    const float* X   = (const float*)d_in[1];   // outputs (N,D) f32
    float* out = (float*)d_out;

    char* ws      = (char*)d_ws;
    double* sumL2 = (double*)(ws + 0);
    double* S     = (double*)(ws + 8);
    int*    n0    = (int*)(ws + 16);
    float*  ibs   = (float*)(ws + 32);               // 5 floats
    float*  sq    = (float*)(ws + 64);               // N floats
    float*  wv    = (float*)(ws + 64 + (size_t)N*4); // N floats

    initk<<<1, 1, 0, stream>>>(sumL2, S, n0);
    rownorm<<<N / 8, 256, 0, stream>>>(X, sub, sq, n0);
    weightsk<<<N / 256, 256, 0, stream>>>(sub, n0, wv);

    dim3 grid(N / BN, N / BM);  // 256 x 128 tiles
    mmd_pass<false><<<grid, 256, 0, stream>>>(X, sq, wv, ibs, sumL2);
    bwk<<<1, 1, 0, stream>>>(sumL2, ibs);
    mmd_pass<true><<<grid, 256, 0, stream>>>(X, sq, wv, ibs, S);
    fin<<<1, 1, 0, stream>>>(S, out);
}